// LowRankGNN_103079215402
// MI455X (gfx1250) — compile-verified
//
#include <hip/hip_runtime.h>

// ---------------- problem constants ----------------
#define NUM_LAYERS 3
#define NUM_BRANCH 4
#define NUM_D      64
#define NUM_M      2048
#define NUM_NN     500000
#define BSZ        20000
#define NF         60000
#define NODES      (BSZ + NF)
#define ESZ        640000
#define CDIM       256      // NUM_BRANCH * NUM_D

typedef __attribute__((ext_vector_type(2))) float v2f;
typedef __attribute__((ext_vector_type(8))) float v8f;
typedef int v4i_t __attribute__((__vector_size__(4 * sizeof(int))));

#define AS1 __attribute__((address_space(1)))
#define AS3 __attribute__((address_space(3)))

#if defined(__has_builtin)
#if __has_builtin(__builtin_amdgcn_global_load_async_to_lds_b128)
#define HAS_ASYNC_LDS 1
#endif
#endif

__device__ __forceinline__ AS3 v4i_t* to_lds_v4(void* p) {
  // generic LDS pointer: low 32 bits are the LDS offset (aperture scheme)
  return (AS3 v4i_t*)(uint32_t)(uintptr_t)p;
}
__device__ __forceinline__ AS1 v4i_t* to_global_v4(const void* p) {
  return (AS1 v4i_t*)(uintptr_t)p;
}
__device__ __forceinline__ void wait_async0() {
#if defined(__has_builtin)
#if __has_builtin(__builtin_amdgcn_s_wait_asynccnt)
  __builtin_amdgcn_s_wait_asynccnt(0);
#else
  asm volatile("s_wait_asynccnt 0x0" ::: "memory");
#endif
#else
  asm volatile("s_wait_asynccnt 0x0" ::: "memory");
#endif
}

// ---------------- kernel 1: build x_input = concat(h, x_fo) ----------------
__global__ __launch_bounds__(256) void build_input_kernel(
    const float* __restrict__ h,            // (BSZ, 256)
    const float* __restrict__ codebooks_l,  // (4, 2048, 128)
    const int*   __restrict__ c_indices_l,  // (4, NUM_NN)
    const int*   __restrict__ first_order_idx, // (NF)
    float* __restrict__ x_input)            // (NODES, 256)
{
  int t = blockIdx.x * 256 + threadIdx.x;     // NODES*64 threads, float4 each
  if (t >= NODES * 64) return;
  int node = t >> 6;
  int q    = t & 63;                          // float4 index within 256-f row
  float4 val;
  if (node < BSZ) {
    val = ((const float4*)h)[node * 64 + q];
  } else {
    int n   = node - BSZ;
    int br  = q >> 4;                         // branch 0..3
    int d4  = q & 15;                         // float4 within first 64 dims
    int fo  = first_order_idx[n];
    int idx = c_indices_l[br * NUM_NN + fo];
    const float* src = codebooks_l + ((size_t)(br * NUM_M + idx)) * (2 * NUM_D);
    val = ((const float4*)src)[d4];           // take [..., :NUM_D]
  }
  ((float4*)x_input)[node * 64 + q] = val;
}

// ---------------- kernel 2: agg = broadcast(bc) ----------------
__global__ __launch_bounds__(256) void init_agg_kernel(
    const float* __restrict__ bc_l, float* __restrict__ agg)
{
  int t = blockIdx.x * 256 + threadIdx.x;     // NODES*64
  if (t >= NODES * 64) return;
  int q = t & 63;
  ((float4*)agg)[t] = ((const float4*)bc_l)[q];
}

// ---------------- kernel 3: FP32 WMMA GEMM  Y = X@W (+bias) (+=Y) (relu) ----
// X: (nTiles*16, 256), W: (256,256) row-major staged whole in LDS (256 KB),
// Y: (nTiles*16, 256). 8 waves; each wave owns a 32-col slab (2x 16-col tiles).
#define TILES_PER_BLOCK 8

__global__ __launch_bounds__(256) void gemm_wmma_kernel(
    const float* __restrict__ X, const float* __restrict__ W,
    const float* __restrict__ bias, float* __restrict__ Y,
    int nTiles, int accumulate, int relu)
{
  extern __shared__ float Wlds[];             // CDIM*CDIM floats = 256 KB
  const int tid  = threadIdx.x;
  const int wid  = tid >> 5;                  // wave 0..7 -> 32-col slab
  const int lane = tid & 31;
  const int half = lane >> 4;                 // 0 | 1
  const int l16  = lane & 15;
  const int tile0   = blockIdx.x * TILES_PER_BLOCK;
  const int tileEnd = (tile0 + TILES_PER_BLOCK < nTiles) ? tile0 + TILES_PER_BLOCK
                                                         : nTiles;
  const int c0 = (wid << 5) + l16;            // first col tile
  const int c1 = c0 + 16;                     // second col tile

  // --- stage ALL of W into LDS once (async DMA, no VGPR round-trip) ---
#ifdef HAS_ASYNC_LDS
#pragma unroll
  for (int i = 0; i < (CDIM * CDIM / 4) / 256; ++i) {  // 64 x b128 per thread
    const int idx4 = tid + i * 256;
    __builtin_amdgcn_global_load_async_to_lds_b128(
        to_global_v4(W + (size_t)idx4 * 4),
        to_lds_v4(Wlds + (size_t)idx4 * 4), 0, 0);
  }
  wait_async0();
#else
#pragma unroll
  for (int i = 0; i < (CDIM * CDIM / 4) / 256; ++i)
    ((float4*)Wlds)[tid + i * 256] = ((const float4*)W)[tid + i * 256];
#endif
  __syncthreads();

  for (int rt = tile0; rt < tileEnd; ++rt) {
    const int r0 = rt << 4;
    // --- accumulator init: bias (+ existing Y when accumulating) ---
    v8f acc0, acc1;
    {
      const float b0 = bias ? bias[c0] : 0.0f;
      const float b1 = bias ? bias[c1] : 0.0f;
#pragma unroll
      for (int v = 0; v < 8; ++v) {
        const int row = r0 + (half << 3) + v; // C/D layout: vgpr v, lane half
        float i0 = b0, i1 = b1;
        if (accumulate) {
          i0 += Y[row * CDIM + c0];
          i1 += Y[row * CDIM + c1];
        }
        acc0[v] = i0;
        acc1[v] = i1;
      }
    }
    // --- full K loop straight out of LDS, no barriers ---
#pragma unroll 4
    for (int kb = 0; kb < CDIM; kb += 4) {
      // A frag (16x4 f32): lanes 0-15 hold K=kb+0,1 ; lanes 16-31 K=kb+2,3
      const int kk = half << 1;
      const float* ap = X + (size_t)(r0 + l16) * CDIM + kb + kk;
      v2f a; a.x = ap[0]; a.y = ap[1];
      // B frags (4x16 f32) from LDS, row stride CDIM
      const float* bp0 = &Wlds[(kb + kk) * CDIM + c0];
      v2f b0; b0.x = bp0[0]; b0.y = bp0[CDIM];
      acc0 = __builtin_amdgcn_wmma_f32_16x16x4_f32(
          false, a, false, b0, (short)0, acc0, false, false);
      const float* bp1 = bp0 + 16;
      v2f b1; b1.x = bp1[0]; b1.y = bp1[CDIM];
      acc1 = __builtin_amdgcn_wmma_f32_16x16x4_f32(
          false, a, false, b1, (short)0, acc1, false, false);
    }
    // --- epilogue ---
#pragma unroll
    for (int v = 0; v < 8; ++v) {
      const int row = r0 + (half << 3) + v;
      float o0 = acc0[v], o1 = acc1[v];
      if (relu) { o0 = fmaxf(o0, 0.0f); o1 = fmaxf(o1, 0.0f); }
      Y[row * CDIM + c0] = o0;
      Y[row * CDIM + c1] = o1;
    }
  }
}

// ---------------- kernel 4: edge scatter  agg[dst] += xw[src]*w ------------
__global__ __launch_bounds__(256) void scatter_kernel(
    const float* __restrict__ xw, const float* __restrict__ ew,
    const int* __restrict__ src, const int* __restrict__ dst,
    float* __restrict__ agg)
{
  int t = blockIdx.x * 256 + threadIdx.x;     // ESZ*64 threads
  if (t >= ESZ * 64) return;
  int e = t >> 6;
  int q = t & 63;
  float w = ew[e];
  int s = src[e], d = dst[e];
  float4 v = ((const float4*)xw)[s * 64 + q];
  float* p = agg + (size_t)d * CDIM + q * 4;
  unsafeAtomicAdd(p + 0, v.x * w);            // global_atomic_add_f32
  unsafeAtomicAdd(p + 1, v.y * w);
  unsafeAtomicAdd(p + 2, v.z * w);
  unsafeAtomicAdd(p + 3, v.w * w);
}

// ---------------- host orchestration ----------------
extern "C" void kernel_launch(void* const* d_in, const int* in_sizes, int n_in,
                              void* d_out, int out_size, void* d_ws, size_t ws_size,
                              hipStream_t stream) {
  const float* x           = (const float*)d_in[0];
  const float* edge_weight = (const float*)d_in[1];
  const float* codebooks   = (const float*)d_in[2];
  const float* Wc          = (const float*)d_in[3];
  const float* bc          = (const float*)d_in[4];
  const float* Wt          = (const float*)d_in[5];
  const float* bt          = (const float*)d_in[6];
  const float* Ws          = (const float*)d_in[7];
  const float* bs          = (const float*)d_in[8];
  const float* Wf          = (const float*)d_in[9];
  const float* bf          = (const float*)d_in[10];
  const int* first_order_idx = (const int*)d_in[11];
  const int* edge_src        = (const int*)d_in[12];
  const int* edge_dst        = (const int*)d_in[13];
  const int* c_indices       = (const int*)d_in[14];
  float* out = (float*)d_out;

  float* ws_f    = (float*)d_ws;
  float* x_input = ws_f;                                   // NODES*256
  float* xw      = x_input + (size_t)NODES * CDIM;         // NODES*256
  float* agg     = xw      + (size_t)NODES * CDIM;         // NODES*256
  float* hA      = agg     + (size_t)NODES * CDIM;         // BSZ*256
  float* hB      = hA      + (size_t)BSZ   * CDIM;         // BSZ*256

  const int gNodesVec = (NODES * 64 + 255) / 256;          // 20000 blocks
  const int gScatter  = (ESZ * 64 + 255) / 256;            // 160000 blocks
  const int tilesAll  = NODES / 16;                        // 5000
  const int tilesB    = BSZ / 16;                          // 1250
  const int gGemmAll  = (tilesAll + TILES_PER_BLOCK - 1) / TILES_PER_BLOCK;
  const int gGemmB    = (tilesB   + TILES_PER_BLOCK - 1) / TILES_PER_BLOCK;
  const size_t WBYTES = (size_t)CDIM * CDIM * sizeof(float);  // 256 KB dyn LDS

  const float* h  = x;
  float*       hN = hA;
  for (int l = 0; l < NUM_LAYERS; ++l) {
    const float* Wc_l = Wc + (size_t)l * CDIM * CDIM;
    const float* bc_l = bc + (size_t)l * CDIM;
    const float* Wt_l = Wt + (size_t)l * CDIM * CDIM;
    const float* bt_l = bt + (size_t)l * CDIM;
    const float* Ws_l = Ws + (size_t)l * CDIM * CDIM;
    const float* bs_l = bs + (size_t)l * CDIM;
    const float* cb_l = codebooks + (size_t)l * NUM_BRANCH * NUM_M * (2 * NUM_D);
    const int*   ci_l = c_indices + (size_t)l * NUM_BRANCH * NUM_NN;

    build_input_kernel<<<gNodesVec, 256, 0, stream>>>(
        h, cb_l, ci_l, first_order_idx, x_input);
    init_agg_kernel<<<gNodesVec, 256, 0, stream>>>(bc_l, agg);
    gemm_wmma_kernel<<<gGemmAll, 256, WBYTES, stream>>>(
        x_input, Wc_l, nullptr, xw, tilesAll, 0, 0);
    scatter_kernel<<<gScatter, 256, 0, stream>>>(
        xw, edge_weight, edge_src, edge_dst, agg);
    // out = agg[:B]@Wt + bt, then += h@Ws + bs, relu except last layer
    gemm_wmma_kernel<<<gGemmB, 256, WBYTES, stream>>>(
        agg, Wt_l, bt_l, hN, tilesB, 0, 0);
    gemm_wmma_kernel<<<gGemmB, 256, WBYTES, stream>>>(
        h, Ws_l, bs_l, hN, tilesB, 1, (l < NUM_LAYERS - 1) ? 1 : 0);
    h  = hN;
    hN = (hN == hA) ? hB : hA;
  }
  gemm_wmma_kernel<<<gGemmB, 256, WBYTES, stream>>>(
      h, Wf, bf, out, tilesB, 0, 0);
}